// PointerGeneratorOut_11184094838999
// MI455X (gfx1250) — compile-verified
//
#include <hip/hip_runtime.h>
#include <stdint.h>

// Problem dims (fixed by the reference)
#define B_     2048
#define VGEN_  50000
#define VINP_  32000
#define S_     200
#define VOUT_  50304

#define NTHREADS 1024
#define CHUNK    (NTHREADS * 4)        // 4096 floats = 16 KB per async chunk
#define NCHUNK   (VGEN_ / CHUNK)       // 12 full chunks
#define TAILBASE (NCHUNK * CHUNK)      // 49152 (848-element tail handled directly)

typedef float v4f __attribute__((ext_vector_type(4)));   // clang-native vector for builtins

// ---- CDNA5 async global -> LDS copy (ASYNCcnt path), one b128 per lane ----
__device__ __forceinline__ void async_copy16(unsigned lds_byte_addr, const float* gsrc) {
    asm volatile("global_load_async_to_lds_b128 %0, %1, off"
                 :: "v"(lds_byte_addr), "v"(gsrc)
                 : "memory");
}
__device__ __forceinline__ void wait_async_le1() {
    asm volatile("s_wait_asynccnt 0x1" ::: "memory");
}
__device__ __forceinline__ void wait_async_0() {
    asm volatile("s_wait_asynccnt 0x0" ::: "memory");
}

// Block-wide online softmax reduction (max m, rescaled sum s). wave32.
__device__ __forceinline__ void block_online_reduce(float& m, float& s,
                                                    float* wm, float* ws) {
    __syncthreads();   // protect scratch reuse across calls
    #pragma unroll
    for (int off = 16; off > 0; off >>= 1) {
        float om = __shfl_xor(m, off);
        float os = __shfl_xor(s, off);
        float nm = fmaxf(m, om);
        s = s * __expf(m - nm) + os * __expf(om - nm);
        m = nm;
    }
    const int lane = threadIdx.x & 31;
    const int wave = threadIdx.x >> 5;
    if (lane == 0) { wm[wave] = m; ws[wave] = s; }
    __syncthreads();
    if (wave == 0) {
        m = wm[lane];          // exactly 32 waves for blockDim 1024
        s = ws[lane];
        #pragma unroll
        for (int off = 16; off > 0; off >>= 1) {
            float om = __shfl_xor(m, off);
            float os = __shfl_xor(s, off);
            float nm = fmaxf(m, om);
            s = s * __expf(m - nm) + os * __expf(om - nm);
            m = nm;
        }
        if (lane == 0) { wm[0] = m; ws[0] = s; }
    }
    __syncthreads();
    m = wm[0];
    s = ws[0];
}

__global__ __launch_bounds__(NTHREADS, 1)
void pointer_generator_out_kernel(const float* __restrict__ gen_scores,
                                  const float* __restrict__ scores,
                                  const float* __restrict__ interp,
                                  const float* __restrict__ gen_zero_mask,
                                  const int*  __restrict__ gen_to_out,
                                  const int*  __restrict__ inp_to_out,
                                  const int*  __restrict__ ctx_inp,
                                  float* __restrict__ out)
{
    // 201216 B accumulator + 32 KB double-buffer stage + scratch  (~234 KB of the 320 KB WGP LDS)
    __shared__ float acc[VOUT_];
    __shared__ float stage[2][CHUNK];
    __shared__ float wm[32];
    __shared__ float ws[32];

    const int tid = threadIdx.x;
    const int b   = blockIdx.x;
    const float* row = gen_scores + (size_t)b * VGEN_;

    // zero the output-row accumulator (barrier inside first reduction covers this)
    for (int i = tid; i < VOUT_; i += NTHREADS) acc[i] = 0.0f;

    // per-lane LDS byte addresses of this thread's 16B slot in each stage buffer
    const unsigned sb0 = (unsigned)(uintptr_t)&stage[0][0] + (unsigned)(tid * 16);
    const unsigned sb1 = (unsigned)(uintptr_t)&stage[1][0] + (unsigned)(tid * 16);

    // ---------------- pass 1: online max / rescaled-sum over the gen row ----------------
    float m = -3.0e38f, s = 0.0f;
    async_copy16(sb0, row + tid * 4);
    for (int c = 0; c < NCHUNK; ++c) {
        if (c + 1 < NCHUNK) {
            async_copy16(((c + 1) & 1) ? sb1 : sb0, row + (c + 1) * CHUNK + tid * 4);
            wait_async_le1();          // chunk c has landed (in-order completion)
        } else {
            wait_async_0();
        }
        const v4f x = *(const v4f*)&stage[c & 1][tid * 4];
        float lm = fmaxf(fmaxf(x.x, x.y), fmaxf(x.z, x.w));
        float nm = fmaxf(m, lm);
        s = s * __expf(m - nm)
          + __expf(x.x - nm) + __expf(x.y - nm) + __expf(x.z - nm) + __expf(x.w - nm);
        m = nm;
    }
    for (int i = TAILBASE + tid; i < VGEN_; i += NTHREADS) {
        float x = row[i];
        float nm = fmaxf(m, x);
        s = s * __expf(m - nm) + __expf(x - nm);
        m = nm;
    }
    block_online_reduce(m, s, wm, ws);
    const float rowmax = m;
    const float itp    = interp[b];
    const float gscale = itp / s;      // interp folded into gen-prob scale

    // ---------------- pass 2: scatter gen probs into LDS accumulator ----------------
    async_copy16(sb0, row + tid * 4);
    for (int c = 0; c < NCHUNK; ++c) {
        if (c + 1 < NCHUNK) {
            async_copy16(((c + 1) & 1) ? sb1 : sb0, row + (c + 1) * CHUNK + tid * 4);
            wait_async_le1();
        } else {
            wait_async_0();
        }
        const int i0 = c * CHUNK + tid * 4;
        const v4f  x = *(const v4f*)&stage[c & 1][tid * 4];
        const int4 o = *(const int4*)&gen_to_out[i0];
        const v4f  z = *(const v4f*)&gen_zero_mask[i0];
        atomicAdd(&acc[o.x], __expf(x.x - rowmax) * gscale * z.x);
        atomicAdd(&acc[o.y], __expf(x.y - rowmax) * gscale * z.y);
        atomicAdd(&acc[o.z], __expf(x.z - rowmax) * gscale * z.z);
        atomicAdd(&acc[o.w], __expf(x.w - rowmax) * gscale * z.w);
    }
    for (int i = TAILBASE + tid; i < VGEN_; i += NTHREADS) {
        atomicAdd(&acc[gen_to_out[i]], __expf(row[i] - rowmax) * gscale * gen_zero_mask[i]);
    }

    // ---------------- pointer / copy distribution (S = 200) ----------------
    const float px = (tid < S_) ? scores[(size_t)b * S_ + tid] : -3.0e38f;
    float pm = px;
    float psum = (tid < S_) ? 1.0f : 0.0f;
    block_online_reduce(pm, psum, wm, ws);
    if (tid < S_) {
        const float alpha = __expf(px - pm) * ((1.0f - itp) / psum);
        const int   o     = inp_to_out[ctx_inp[(size_t)b * S_ + tid]];
        atomicAdd(&acc[o], alpha);
    }
    __syncthreads();   // all LDS atomics done before streaming out

    // ---------------- write the finished output row (non-temporal b128) ----------------
    float* orow = out + (size_t)b * VOUT_;
    for (int i = tid * 4; i < VOUT_; i += NTHREADS * 4) {
        const v4f v = *(const v4f*)&acc[i];
        __builtin_nontemporal_store(v, (v4f*)&orow[i]);
    }
}

extern "C" void kernel_launch(void* const* d_in, const int* in_sizes, int n_in,
                              void* d_out, int out_size, void* d_ws, size_t ws_size,
                              hipStream_t stream) {
    const float* gen_scores    = (const float*)d_in[0];
    const float* scores        = (const float*)d_in[1];
    const float* interp        = (const float*)d_in[2];
    const float* gen_zero_mask = (const float*)d_in[3];
    const int*   gen_to_out    = (const int*)  d_in[4];
    const int*   inp_to_out    = (const int*)  d_in[5];
    const int*   ctx_inp       = (const int*)  d_in[6];
    float* out = (float*)d_out;

    pointer_generator_out_kernel<<<dim3(B_), dim3(NTHREADS), 0, stream>>>(
        gen_scores, scores, interp, gen_zero_mask, gen_to_out, inp_to_out, ctx_inp, out);
}